// ContextEncoderEMA_9122510536766
// MI455X (gfx1250) — compile-verified
//
#include <hip/hip_runtime.h>

// ContextEncoderEMA on MI455X (gfx1250, wave32).
// Bandwidth-bound (~452 MB one-pass => ~19us @ 23.3 TB/s). Weighted segment
// sum expressed as W[16,K] x E[K,768] per group of 16 contiguous dialogues,
// accumulated on the matrix pipe with V_WMMA_F32_16X16X4_F32.

#define DIM      768
#define OUTD     1536
#define TAUF     0.9f
#define GROUP    16     // dialogues per group (WMMA M)
#define WSTRIDE  520    // LDS weight-row stride in floats (bank-friendly: 520%64=8)

typedef float v2f __attribute__((ext_vector_type(2)));
typedef float v4f __attribute__((ext_vector_type(4)));
typedef float v8f __attribute__((ext_vector_type(8)));

// ---------------------------------------------------------------------------
// Exclusive prefix sum of lens into starts[0..n], starts[n] = total.
// Single block, 256 threads, Hillis-Steele over per-thread chunk sums.
// ---------------------------------------------------------------------------
__global__ __launch_bounds__(256) void scan_kernel(const int* __restrict__ lens,
                                                   int n,
                                                   int* __restrict__ starts) {
    __shared__ int sums[256];
    const int t = threadIdx.x;
    const int chunk = (n + 255) / 256;
    const int b = t * chunk;
    const int e = min(b + chunk, n);

    int s = 0;
    for (int i = b; i < e; ++i) s += lens[i];
    sums[t] = s;

    for (int off = 1; off < 256; off <<= 1) {
        __syncthreads();
        int v = (t >= off) ? sums[t - off] : 0;
        __syncthreads();
        sums[t] += v;
    }
    __syncthreads();

    int run = (t == 0) ? 0 : sums[t - 1];   // exclusive prefix at chunk start
    for (int i = b; i < e; ++i) { starts[i] = run; run += lens[i]; }
    if (t == 255) starts[n] = run;          // total (thread 255's chunk ends at n)
}

// ---------------------------------------------------------------------------
// Main kernel: one block per group of 16 dialogues.
//   Phase 1: build 16xK weight matrix in LDS (EMA closed-form weights).
//   Phase 2: 8 waves; wave w handles N-tiles w, w+8, ... of 48; each tile is
//            C[16,16] += A[16,4] x B[4,16] chained over K via WMMA f32 16x16x4.
//            Fast path: pointer-strided B loads, no clamps, unguarded stores.
//   Phase 3: 128-bit copy of each dialogue's last utterance to out[:, 768:].
// ---------------------------------------------------------------------------
__global__ __launch_bounds__(256) void ema_wmma_kernel(const float* __restrict__ E,
                                                       const int* __restrict__ lens,
                                                       const int* __restrict__ starts,
                                                       float* __restrict__ out,
                                                       int n_dias,
                                                       int total_rows) {
    __shared__ float W[GROUP * WSTRIDE];     // 33,280 B of 320 KB

    const int g     = blockIdx.x;
    const int d0    = g * GROUP;
    const int count = min(GROUP, n_dias - d0);
    const int base  = starts[d0];
    const int kend  = starts[min(d0 + GROUP, n_dias)];
    const int K_total = kend - base;

    const int tid = threadIdx.x;

    // ---- Phase 1: weights in LDS --------------------------------------
    for (int i = tid; i < GROUP * WSTRIDE; i += 256) W[i] = 0.0f;
    __syncthreads();

    if (tid < count) {
        const int m = tid;
        const int s = starts[d0 + m] - base;   // local row offset of dialogue m
        const int L = lens[d0 + m];
        for (int k = 0; k < L; ++k) {
            float w;
            if (k == L - 1)      w = 0.0f;                                   // last excluded
            else if (k == L - 2) w = __powf(TAUF, (float)(L - 2));           // seed term
            else                 w = (1.0f - TAUF) * __powf(TAUF, (float)k); // recurrence
            W[m * WSTRIDE + s + k] = w;
        }
    }
    __syncthreads();

    // ---- Phase 2: WMMA accumulation -----------------------------------
    const int wave = tid >> 5;        // 0..7
    const int lane = tid & 31;
    const int half = lane >> 4;       // 0: K pair {k,k+1}; 1: K pair {k+2,k+3}
    const int nl   = lane & 15;       // M index (for A) / N index (for B,C)
    const int Ksteps = (K_total + 3) >> 2;

    // Uniform per block: padded K-range stays inside the table for all but
    // the final group(s); same for full-vs-partial dialogue count.
    const bool in_bounds = (kend + 3 <= total_rows);
    const bool full      = (count == GROUP);

    for (int nt = wave; nt < DIM / 16; nt += 8) {
        const int n0 = nt * 16;
        v8f c = {};                   // f32 accumulator, 8 VGPRs

        if (in_bounds) {
            // Fast path: per-lane pointer strided by 4 rows per step.
            const float* p = E + (size_t)(base + half * 2) * DIM + n0 + nl;
            const float* wp = &W[nl * WSTRIDE + half * 2];
            for (int ks = 0; ks < Ksteps; ++ks) {
                v2f a;
                a.x = wp[0];
                a.y = wp[1];
                v2f b;
                b.x = p[0];
                b.y = p[DIM];
                p  += 4 * DIM;
                wp += 4;
                // (neg_a, A, neg_b, B, c_mod, C, reuse_a, reuse_b)
                c = __builtin_amdgcn_wmma_f32_16x16x4_f32(false, a, false, b,
                                                          (short)0, c, false, false);
            }
        } else {
            // Tail path: clamp row indices (weights are zero in the overhang).
            for (int ks = 0; ks < Ksteps; ++ks) {
                const int ak = ks * 4 + half * 2;
                v2f a;
                a.x = W[nl * WSTRIDE + ak];
                a.y = W[nl * WSTRIDE + ak + 1];
                int r0 = min(base + ak,     total_rows - 1);
                int r1 = min(base + ak + 1, total_rows - 1);
                v2f b;
                b.x = E[(size_t)r0 * DIM + n0 + nl];
                b.y = E[(size_t)r1 * DIM + n0 + nl];
                c = __builtin_amdgcn_wmma_f32_16x16x4_f32(false, a, false, b,
                                                          (short)0, c, false, false);
            }
        }

        // C/D layout: VGPR j -> M=j (lanes 0-15) or M=8+j (lanes 16-31), N=nl
        if (full) {
            #pragma unroll
            for (int j = 0; j < 8; ++j) {
                const int M = j + half * 8;
                __builtin_nontemporal_store(
                    c[j], &out[(size_t)(d0 + M) * OUTD + n0 + nl]);
            }
        } else {
            #pragma unroll
            for (int j = 0; j < 8; ++j) {
                const int M = j + half * 8;
                if (M < count)
                    __builtin_nontemporal_store(
                        c[j], &out[(size_t)(d0 + M) * OUTD + n0 + nl]);
            }
        }
    }

    // ---- Phase 3: copy last utterance (128-bit streaming) -------------
    const int F4 = DIM / 4;           // 192 x 16B per row
    const v4f* __restrict__ E4 = (const v4f*)E;
    v4f* __restrict__ out4 = (v4f*)out;
    for (int idx = tid; idx < count * F4; idx += 256) {
        const int m  = idx / F4;
        const int c4 = idx - m * F4;
        const int last_row = starts[d0 + m] + lens[d0 + m] - 1;
        v4f v = E4[(size_t)last_row * F4 + c4];
        __builtin_nontemporal_store(
            v, &out4[(size_t)(d0 + m) * (OUTD / 4) + F4 + c4]);
    }
}

// ---------------------------------------------------------------------------
extern "C" void kernel_launch(void* const* d_in, const int* in_sizes, int n_in,
                              void* d_out, int out_size, void* d_ws, size_t ws_size,
                              hipStream_t stream) {
    const float* E    = (const float*)d_in[0];   // [total, 768] f32
    const int*   lens = (const int*)d_in[1];     // [n_dias] i32
    const int n_dias     = in_sizes[1];
    const int total_rows = in_sizes[0] / DIM;

    int* starts = (int*)d_ws;                    // (n_dias + 1) ints

    scan_kernel<<<1, 256, 0, stream>>>(lens, n_dias, starts);

    const int groups = (n_dias + GROUP - 1) / GROUP;
    ema_wmma_kernel<<<groups, 256, 0, stream>>>(E, lens, starts,
                                                (float*)d_out, n_dias, total_rows);
}